// BidirectionalCursedIIR_89249420410955
// MI455X (gfx1250) — compile-verified
//
#include <hip/hip_runtime.h>
#include <hip/hip_bf16.h>

// ---------------------------------------------------------------------------
// Bidirectional 64-channel biquad resonator bank on MI455X (gfx1250).
//
// Algorithm: block decomposition of the 2nd-order linear recurrence.
//   y[t] = b0*x[t] - a1*y[t-1] - a2*y[t-2]
// Split T into blocks of L=16. For each block:
//   y[m] = yp[m] + alpha*g1[m] + beta*g2[m]
// where yp = H_c * x_block (16x16 lower-triangular Toeplitz of the impulse
// response -> WMMA f32 16x16x4), g1/g2 = zero-input responses to unit initial
// states (f64), and (alpha,beta) = incoming (y[-1],y[-2]) carried across
// blocks in FULL f64 (phase 1 computes per-block zero-state end states in
// f64; phase 2 scans a 2x2 f64 transition per chain). The f32 WMMA error is
// therefore purely local (no feedback) -> output accurate to f32 rounding.
// ---------------------------------------------------------------------------

typedef __attribute__((ext_vector_type(2))) float v2f;
typedef __attribute__((ext_vector_type(8))) float v8f;

namespace {
constexpr int B       = 8;
constexpr int C       = 64;
constexpr int T       = 24000;
constexpr int L       = 16;            // block length == WMMA tile M
constexpr int NBLK    = T / L;         // 1500 blocks per chain
constexpr int NCHAIN  = 2 * B * C;     // dir x batch x channel = 1024
constexpr int NTGRP   = (NBLK + 15) / 16;  // 94 tile groups (16 blocks each)

// workspace layout (bytes)
constexpr size_t OFF_H = 0;                       // f32 [C][4][32][2] packed WMMA-A
constexpr size_t SZ_H  = (size_t)C * 4 * 32 * 2 * sizeof(float);   // 64 KB
constexpr size_t OFF_G = OFF_H + SZ_H;            // f64 [C][2][16]  g1,g2
constexpr size_t SZ_G  = (size_t)C * 2 * 16 * sizeof(double);      // 16 KB
constexpr size_t OFF_M = OFF_G + SZ_G;            // f64 [C][4]      2x2 block transition
constexpr size_t SZ_M  = (size_t)C * 4 * sizeof(double);           // 2 KB
constexpr size_t OFF_S = (OFF_M + SZ_M + 255) & ~(size_t)255;      // f64 [chain][blk][2]
constexpr size_t SZ_S  = (size_t)NCHAIN * NBLK * 2 * sizeof(double); // 24.58 MB
} // namespace

// ---------------------------------------------------------------------------
// Phase 0: per-channel tables. h = impulse response, g1/g2 = zero-input
// responses to (y-1,y-2)=(1,0)/(0,1). H packed in the exact VGPR lane layout
// of the 32-bit 16x4 WMMA A operand (ISA 7.12.2): chunk q covers K=4q..4q+3;
// lanes 0-15 hold (K=4q, 4q+1) of row M=lane, lanes 16-31 hold (4q+2, 4q+3).
// ---------------------------------------------------------------------------
__global__ void __launch_bounds__(64) iir_setup(const double* __restrict__ a,
                                                const double* __restrict__ bc,
                                                float* __restrict__ H,
                                                double* __restrict__ G,
                                                double* __restrict__ M) {
  int c = threadIdx.x;
  if (c >= C) return;
  double a1 = a[c * 3 + 1], a2 = a[c * 3 + 2], b0 = bc[c * 3 + 0];

  double h[L], g1[L], g2[L];
  h[0]  = b0;        h[1]  = -a1 * h[0];
  g1[0] = -a1;       g1[1] = -a1 * g1[0] - a2;   // y-1 = 1
  g2[0] = -a2;       g2[1] = -a1 * g2[0];        // y-2 = 1
  for (int t = 2; t < L; ++t) {
    h[t]  = -a1 * h[t - 1]  - a2 * h[t - 2];
    g1[t] = -a1 * g1[t - 1] - a2 * g1[t - 2];
    g2[t] = -a1 * g2[t - 1] - a2 * g2[t - 2];
  }
  for (int t = 0; t < L; ++t) {
    G[(size_t)(c * 2 + 0) * L + t] = g1[t];
    G[(size_t)(c * 2 + 1) * L + t] = g2[t];
  }
  // block state transition:  (y[15], y[14]) = M * (y-1, y-2) + (p15, p14)
  M[c * 4 + 0] = g1[15]; M[c * 4 + 1] = g2[15];
  M[c * 4 + 2] = g1[14]; M[c * 4 + 3] = g2[14];

  for (int q = 0; q < 4; ++q)
    for (int lane = 0; lane < 32; ++lane) {
      int m  = lane & 15;
      int k0 = q * 4 + (lane >> 4) * 2;
      for (int j = 0; j < 2; ++j) {
        int k = k0 + j;
        float v = (m >= k) ? (float)h[m - k] : 0.0f;   // Toeplitz H[m][k]=h[m-k]
        H[(((size_t)c * 4 + q) * 32 + lane) * 2 + j] = v;
      }
    }
}

// ---------------------------------------------------------------------------
// Phase 1: per-block zero-state end states (yp[15], yp[14]) in f64.
// 1.536M independent threads x 16 serial steps -> the 24000-long critical
// path collapses to 16 dependent f64 FMAs here.
// ---------------------------------------------------------------------------
__global__ void __launch_bounds__(256) iir_pstates(const float* __restrict__ x,
                                                   const double* __restrict__ a,
                                                   const double* __restrict__ bc,
                                                   double* __restrict__ S) {
  int gid = blockIdx.x * blockDim.x + threadIdx.x;
  if (gid >= NCHAIN * NBLK) return;
  int blk   = gid % NBLK;
  int chain = gid / NBLK;
  int c = chain & 63, b = (chain >> 6) & 7, dir = chain >> 9;

  double a1 = a[c * 3 + 1], a2 = a[c * 3 + 2], b0 = bc[c * 3 + 0];
  const float* xb = x + (size_t)b * T;

  double y1 = 0.0, y2 = 0.0;
  int s0 = blk * L;
  #pragma unroll
  for (int i = 0; i < L; ++i) {
    int s = s0 + i;
    int t = dir ? (T - 1 - s) : s;
    double xv = (double)xb[t];
    double y  = b0 * xv - a1 * y1 - a2 * y2;
    y2 = y1; y1 = y;
  }
  S[(size_t)gid * 2 + 0] = y1;   // yp[15]
  S[(size_t)gid * 2 + 1] = y2;   // yp[14]
}

// ---------------------------------------------------------------------------
// Phase 2: sequential f64 carry scan per chain (1024 lanes, 1500 steps of a
// 2x2 matvec). Overwrites each block's slot with its INCOMING (alpha,beta).
// ---------------------------------------------------------------------------
__global__ void __launch_bounds__(256) iir_scan(const double* __restrict__ M,
                                                double* __restrict__ S) {
  int chain = blockIdx.x * blockDim.x + threadIdx.x;
  if (chain >= NCHAIN) return;
  int c = chain & 63;
  double m00 = M[c * 4 + 0], m01 = M[c * 4 + 1];
  double m10 = M[c * 4 + 2], m11 = M[c * 4 + 3];

  double s1 = 0.0, s2 = 0.0;  // incoming (y-1, y-2)
  double* Sc = S + (size_t)chain * NBLK * 2;
  for (int k = 0; k < NBLK; ++k) {
    double p1 = Sc[k * 2 + 0], p2 = Sc[k * 2 + 1];
    Sc[k * 2 + 0] = s1;               // alpha for block k
    Sc[k * 2 + 1] = s2;               // beta  for block k
    double n1 = p1 + m00 * s1 + m01 * s2;
    double n2 = p2 + m10 * s1 + m11 * s2;
    s1 = n1; s2 = n2;
  }
}

// ---------------------------------------------------------------------------
// Phase 3: one wave per 16x16 tile = (one chain, 16 consecutive blocks).
//   yp tile = H_c(16x16) x X(16x16)  via 4 chained v_wmma_f32_16x16x4_f32
//   y       = yp + alpha*g1[m] + beta*g2[m]   (correction in f64)
// Tile is transposed through LDS so global stores are contiguous float4s
// (reversed float4s for the backward direction).
// ---------------------------------------------------------------------------
__global__ void __launch_bounds__(256) iir_reconstruct(const float* __restrict__ x,
                                                       const float* __restrict__ H,
                                                       const double* __restrict__ G,
                                                       const double* __restrict__ S,
                                                       float* __restrict__ out) {
  __shared__ float lds_tile[8 * 256];   // 8 waves/block, 16x16 f32 tile each

  int wave = (blockIdx.x * blockDim.x + threadIdx.x) >> 5;
  int lane = threadIdx.x & 31;
  bool wvalid = wave < NCHAIN * NTGRP;  // wave-uniform (grid is exact anyway)

  float* tile = lds_tile + ((threadIdx.x >> 5) & 7) * 256;

  int tg = 0, chain = 0, c = 0, b = 0, dir = 0, col = 0, blk = 0;
  bool valid = false;

  if (wvalid) {
    tg    = wave % NTGRP;
    chain = wave / NTGRP;
    c = chain & 63; b = (chain >> 6) & 7; dir = chain >> 9;
    const float* xb = x + (size_t)b * T;

    // A operand: pre-packed per-lane pairs, one b64 load per K-chunk.
    const v2f* Hc = (const v2f*)(H + (size_t)c * 4 * 32 * 2);
    v2f A0 = Hc[0 * 32 + lane];
    v2f A1 = Hc[1 * 32 + lane];
    v2f A2 = Hc[2 * 32 + lane];
    v2f A3 = Hc[3 * 32 + lane];

    // B operand: column n = block, K = sample-in-block. Lane n<16 supplies
    // (K=4q,4q+1), lane n+16 supplies (4q+2,4q+3) of the same column.
    col = lane & 15;
    blk = tg * 16 + col;
    valid = blk < NBLK;
    int kb = (lane >> 4) * 2;
    v2f B0 = {0.0f, 0.0f}, B1 = {0.0f, 0.0f}, B2 = {0.0f, 0.0f}, B3 = {0.0f, 0.0f};
    if (valid) {
      int sbase = blk * L + kb;
      if (!dir) {
        const float* p = xb + sbase;
        B0.x = p[0];   B0.y = p[1];
        B1.x = p[4];   B1.y = p[5];
        B2.x = p[8];   B2.y = p[9];
        B3.x = p[12];  B3.y = p[13];
      } else {
        const float* p = xb + (T - 1) - sbase;   // reversed input
        B0.x = p[0];   B0.y = p[-1];
        B1.x = p[-4];  B1.y = p[-5];
        B2.x = p[-8];  B2.y = p[-9];
        B3.x = p[-12]; B3.y = p[-13];
      }
    }

    // EXEC is all-ones here (reconverged): 4 chained WMMAs accumulate K=16.
    v8f acc = {};
    acc = __builtin_amdgcn_wmma_f32_16x16x4_f32(false, A0, false, B0, (short)0, acc, false, false);
    acc = __builtin_amdgcn_wmma_f32_16x16x4_f32(false, A1, false, B1, (short)0, acc, false, false);
    acc = __builtin_amdgcn_wmma_f32_16x16x4_f32(false, A2, false, B2, (short)0, acc, false, false);
    acc = __builtin_amdgcn_wmma_f32_16x16x4_f32(false, A3, false, B3, (short)0, acc, false, false);

    // Homogeneous correction in f64, stage transposed tile into LDS.
    if (valid) {
      double alpha = S[((size_t)chain * NBLK + blk) * 2 + 0];
      double beta  = S[((size_t)chain * NBLK + blk) * 2 + 1];
      const double* g1 = G + (size_t)(c * 2 + 0) * L;
      const double* g2 = G + (size_t)(c * 2 + 1) * L;
      int rb = (lane >> 4) * 8;   // C/D layout: lanes 0-15 rows 0-7, 16-31 rows 8-15
      #pragma unroll
      for (int r = 0; r < 8; ++r) {
        int m = rb + r;
        double yd = (double)acc[r] + alpha * g1[m] + beta * g2[m];
        tile[col * 16 + m] = (float)yd;   // tile index == local time index
      }
    }
  }

  __syncthreads();

  if (wvalid) {
    float* ob = out + ((size_t)b * 128 + (size_t)dir * 64 + c) * T;
    #pragma unroll
    for (int half = 0; half < 2; ++half) {
      int tl = lane * 8 + half * 4;
      int s0 = tg * 256 + tl;
      if (s0 + 3 < T) {
        float4 v = *(const float4*)(tile + tl);
        if (!dir) {
          *(float4*)(ob + s0) = v;                       // contiguous ascending
        } else {
          float4 rv = make_float4(v.w, v.z, v.y, v.x);   // contiguous descending
          *(float4*)(ob + (T - 4 - s0)) = rv;
        }
      } else {
        for (int j = 0; j < 4; ++j) {                    // ragged tail (tg==93)
          int s = s0 + j;
          if (s < T) ob[dir ? (T - 1 - s) : s] = tile[tl + j];
        }
      }
    }
  }
}

// ---------------------------------------------------------------------------
extern "C" void kernel_launch(void* const* d_in, const int* in_sizes, int n_in,
                              void* d_out, int out_size, void* d_ws, size_t ws_size,
                              hipStream_t stream) {
  (void)in_sizes; (void)n_in; (void)out_size; (void)ws_size;
  const float*  x  = (const float*)d_in[0];    // (8,1,24000) f32
  const double* a  = (const double*)d_in[1];   // (64,3) f64
  const double* bc = (const double*)d_in[2];   // (64,3) f64
  float* out = (float*)d_out;                  // (8,128,24000) f32

  char* ws = (char*)d_ws;                      // needs ~24.7 MB
  float*  H = (float*)(ws + OFF_H);
  double* G = (double*)(ws + OFF_G);
  double* M = (double*)(ws + OFF_M);
  double* S = (double*)(ws + OFF_S);

  iir_setup<<<1, 64, 0, stream>>>(a, bc, H, G, M);

  constexpr int P1_THREADS = NCHAIN * NBLK;            // 1,536,000
  iir_pstates<<<P1_THREADS / 256, 256, 0, stream>>>(x, a, bc, S);

  iir_scan<<<NCHAIN / 256, 256, 0, stream>>>(M, S);

  constexpr int P3_WAVES = NCHAIN * NTGRP;             // 96,256
  iir_reconstruct<<<P3_WAVES / 8, 256, 0, stream>>>(x, H, G, S, out);
}